// AtomConvLayer_86242943304061
// MI455X (gfx1250) — compile-verified
//
#include <hip/hip_runtime.h>
#include <hip/hip_bf16.h>

typedef __attribute__((ext_vector_type(2))) float v2f;
typedef __attribute__((ext_vector_type(8))) float v8f;

#define EPSV 1e-12f
#define B_  16
#define N_  1024
#define M_  12
#define FA  128
#define FB  32
#define FO  128

// -------------------------------------------------------------------------
// Kernel 1: per-(b,n) bond-norm chain up to the product over M.
// bond slice per (b,n) is M_*FB = 384 contiguous floats; one wave32 per (b,n).
// bond_norm = (||bond||_2)^-2 = 1/sum(bond^2);  L1-normalize over M then
// product over M  ==  prod(inv) / denom^12.
// -------------------------------------------------------------------------
__global__ void k_bondval(const float* __restrict__ bond, float* __restrict__ val)
{
    int lane = threadIdx.x & 31;
    int wave = threadIdx.x >> 5;
    int bn   = blockIdx.x * (blockDim.x >> 5) + wave;   // 0..B_*N_-1
    const float* p = bond + (size_t)bn * (M_ * FB);

    float prod = 1.0f, denom = 0.0f;
#pragma unroll
    for (int m = 0; m < M_; ++m) {
        float x = p[m * FB + lane];       // FB==32 == wave width
        float s = x * x;
#pragma unroll
        for (int off = 16; off > 0; off >>= 1)
            s += __shfl_xor(s, off, 32);  // full-wave sum, all lanes get it
        float inv = 1.0f / s;             // ||.||^-2
        denom += inv;
        prod  *= inv;
    }
    denom = fmaxf(denom, EPSV);
    float d2 = denom * denom, d4 = d2 * d2, d8 = d4 * d4;
    float d12 = d8 * d4;
    if (lane == 0) val[bn] = prod / d12;
}

// -------------------------------------------------------------------------
// Kernel 2: colsum[n] = sum_b |val[b,n]|   (L1 denom over batch axis)
// -------------------------------------------------------------------------
__global__ void k_colsum(const float* __restrict__ val, float* __restrict__ colsum)
{
    int n = blockIdx.x * blockDim.x + threadIdx.x;
    if (n < N_) {
        float s = 0.0f;
#pragma unroll
        for (int b = 0; b < B_; ++b) s += fabsf(val[b * N_ + n]);
        colsum[n] = s;
    }
}

// -------------------------------------------------------------------------
// Kernel 3: u[b,n] = (val/colsum)^-1 = max(colsum,eps)/val ; rowsum[b] = L1
// over N of u.  One block per batch b.
// -------------------------------------------------------------------------
__global__ void k_rownorm(const float* __restrict__ val,
                          const float* __restrict__ colsum,
                          float* __restrict__ u, float* __restrict__ rowsum)
{
    __shared__ float red[256];
    int b = blockIdx.x;
    float part = 0.0f;
    for (int n = threadIdx.x; n < N_; n += blockDim.x) {
        float t = fmaxf(colsum[n], EPSV) / val[b * N_ + n];
        u[b * N_ + n] = t;
        part += fabsf(t);
    }
    red[threadIdx.x] = part;
    __syncthreads();
    for (int s = blockDim.x >> 1; s > 0; s >>= 1) {
        if (threadIdx.x < (unsigned)s) red[threadIdx.x] += red[threadIdx.x + s];
        __syncthreads();
    }
    if (threadIdx.x == 0) rowsum[b] = fmaxf(red[0], EPSV);
}

// -------------------------------------------------------------------------
// Kernel 4: fused gather + mean + scale + GEMM(128x128) + bias + relu.
// Block = 128 threads (4 waves).  Block computes 64 output rows x 128 cols.
// Phase 1: build X[64][128] tile in LDS (padded stride 132) AND stage W in
//          LDS in *WMMA fragment order*:
//            Wf[((k0/4)*8 + t)*64 + lane*2 + {0,1}]
//              = W[k0 + ko + {0,1}][t*16 + col]
//          so each lane's B fragment is one aligned ds_load_b64, lane l hits
//          banks 2l/2l+1 (conflict-free), and no register repacking is
//          needed in front of v_wmma_f32_16x16x4_f32.
// -------------------------------------------------------------------------
#define XSTRIDE 132

__global__ __launch_bounds__(128) void
k_main(const float* __restrict__ atom, const int* __restrict__ adj,
       const float* __restrict__ W, const float* __restrict__ bias,
       const float* __restrict__ u, const float* __restrict__ rowsum,
       float* __restrict__ out)
{
    __shared__ float Xs[64 * XSTRIDE];     // 33792 B
    __shared__ float Wf[32 * 8 * 64];      // 65536 B (fragment-ordered W)
    int tid = threadIdx.x;
    int rowBase = blockIdx.x * 64;

    // ---- Phase 1a: stage W into LDS in fragment order (2 frags/thread) ----
    {
#pragma unroll
        for (int ff = 0; ff < 2; ++ff) {
            int f  = ff * 128 + tid;         // fragment id 0..255 = (k0/4)*8+t
            int k0 = (f >> 3) << 2;          // K-step base row
            int t  = f & 7;                  // column tile
            const float* w0 = W + (size_t)k0 * FO + t * 16;
            float* dst = &Wf[f * 64];
#pragma unroll
            for (int hh = 0; hh < 2; ++hh) { // ko = 0 or 2 (lane halves)
                const float4* r0 = (const float4*)(w0 + (hh * 2) * FO);
                const float4* r1 = (const float4*)(w0 + (hh * 2 + 1) * FO);
#pragma unroll
                for (int c4 = 0; c4 < 4; ++c4) {
                    float4 a0 = r0[c4];
                    float4 a1 = r1[c4];
                    float2* d = (float2*)&dst[(hh * 16 + c4 * 4) * 2];
                    d[0] = make_float2(a0.x, a1.x);
                    d[1] = make_float2(a0.y, a1.y);
                    d[2] = make_float2(a0.z, a1.z);
                    d[3] = make_float2(a0.w, a1.w);
                }
            }
        }
    }

    // ---- Phase 1b: X tile = (atom + mean(atom[adj])) * scale ----
    {
        int r  = tid & 63;           // row within tile; 2 threads per row
        int hf = tid >> 6;           // 0: cols 0..63, 1: cols 64..127
        int R  = rowBase + r;
        int b  = R >> 10;            // / N_
        int n  = R & (N_ - 1);
        float scale = u[R] / rowsum[b];

        int idx[M_];
        const int* arow = adj + (size_t)R * M_;
#pragma unroll
        for (int m = 0; m < M_; ++m) idx[m] = arow[m];

        const float4* atomB = (const float4*)(atom + (size_t)b * N_ * FA);
        const float4* self  = atomB + (size_t)n * (FA / 4);
        const float inv12 = 1.0f / 12.0f;
#pragma unroll
        for (int j = 0; j < 16; ++j) {
            int c = hf * 16 + j;                     // float4 index in row
            float4 a = self[c];
            float sx = 0.f, sy = 0.f, sz = 0.f, sw = 0.f;
#pragma unroll
            for (int m = 0; m < M_; ++m) {
                float4 nb = atomB[(size_t)idx[m] * (FA / 4) + c];
                sx += nb.x; sy += nb.y; sz += nb.z; sw += nb.w;
            }
            float4 x;
            x.x = (a.x + sx * inv12) * scale;
            x.y = (a.y + sy * inv12) * scale;
            x.z = (a.z + sz * inv12) * scale;
            x.w = (a.w + sw * inv12) * scale;
            *(float4*)&Xs[r * XSTRIDE + c * 4] = x;
        }
    }
    __syncthreads();

    // ---- Phase 2: WMMA f32 16x16x4, 8 column tiles per wave ----
    int lane = tid & 31;
    int wave = tid >> 5;                 // 4 waves -> 4 row stripes of 16
    int ar   = lane & 15;                // A-matrix row within stripe
    int ko   = (lane >> 4) << 1;         // lanes 16..31 hold K+2,K+3

    v8f acc[8];
#pragma unroll
    for (int t = 0; t < 8; ++t) acc[t] = (v8f)0.0f;

    const float* Xrow = &Xs[(wave * 16 + ar) * XSTRIDE];
    const float* wlane = &Wf[lane * 2];
    for (int k0 = 0; k0 < FA; k0 += 4) {
        v2f a;
        a.x = Xrow[k0 + ko];
        a.y = Xrow[k0 + ko + 1];
        const float* wk = wlane + (k0 << 7);   // (k0/4)*8*64
#pragma unroll
        for (int t = 0; t < 8; ++t) {
            v2f bb = *(const v2f*)(wk + t * 64);
            acc[t] = __builtin_amdgcn_wmma_f32_16x16x4_f32(
                false, a, false, bb, (short)0, acc[t], false, false);
        }
    }

    // ---- Epilogue: bias + relu + store (C/D layout: VGPR v -> row v/v+8) ----
    int col   = lane & 15;
    int rhalf = (lane >> 4) * 8;
#pragma unroll
    for (int t = 0; t < 8; ++t) {
        int c = t * 16 + col;
        float bs = bias[c];
#pragma unroll
        for (int v = 0; v < 8; ++v) {
            int row = rowBase + wave * 16 + rhalf + v;
            float r = acc[t][v] + bs;
            out[(size_t)row * FO + c] = fmaxf(r, 0.0f);
        }
    }
}

// -------------------------------------------------------------------------
extern "C" void kernel_launch(void* const* d_in, const int* in_sizes, int n_in,
                              void* d_out, int out_size, void* d_ws, size_t ws_size,
                              hipStream_t stream)
{
    const float* atom = (const float*)d_in[0];   // (B,N,FA)
    const float* bond = (const float*)d_in[1];   // (B,N,M,FB)
    const int*   adj  = (const int*)d_in[2];     // (B,N,M) int32
    const float* W    = (const float*)d_in[3];   // (FA,FO)
    const float* bias = (const float*)d_in[4];   // (FO)
    float* out = (float*)d_out;

    float* ws     = (float*)d_ws;
    float* val    = ws;                  // B_*N_ = 16384
    float* u      = ws + 16384;          // 16384
    float* colsum = ws + 32768;          // 1024
    float* rowsum = ws + 33792;          // 16

    k_bondval<<<(B_ * N_) / 8, 256, 0, stream>>>(bond, val);
    k_colsum <<<(N_ + 255) / 256, 256, 0, stream>>>(val, colsum);
    k_rownorm<<<B_, 256, 0, stream>>>(val, colsum, u, rowsum);
    k_main   <<<(B_ * N_) / 64, 128, 0, stream>>>(atom, adj, W, bias, u, rowsum, out);
}